// DeepONet_41644002902279
// MI455X (gfx1250) — compile-verified
//
#include <hip/hip_runtime.h>
#include <math.h>

typedef __attribute__((ext_vector_type(2))) float v2f;
typedef __attribute__((ext_vector_type(8))) float v8f;

#define BDIM 4096
#define EDIM 4096
#define PW   128       // basis width (K)
#define ROWS 4         // rows per MLP block

#define MAT_STRIDE (BDIM * PW)                 // floats between staged matrices
#define MAT_BYTES  (MAT_STRIDE * 4)            // 2 MB, fits imm24 offset
#define ROWB       (PW * 4)                    // 512 bytes per row

// ---------------------------------------------------------------------------
// Branch MLP (1 -> 128x4, tanh) forward + JVP with tangent = ones.
// Folds w_lin:  A = basis * w_lin,  Amu = jvp * w_lin
// ---------------------------------------------------------------------------
__global__ __launch_bounds__(PW) void branch_mlp_kernel(
    const float* __restrict__ DT,
    const float* __restrict__ W1, const float* __restrict__ B1,
    const float* __restrict__ W2, const float* __restrict__ B2,
    const float* __restrict__ W3, const float* __restrict__ B3,
    const float* __restrict__ W4, const float* __restrict__ B4,
    const float* __restrict__ wlin,
    float* __restrict__ A, float* __restrict__ Amu)
{
    __shared__ float sa[ROWS][PW];
    __shared__ float sd[ROWS][PW];
    const int j    = threadIdx.x;
    const int row0 = blockIdx.x * ROWS;

    float y[ROWS], dy[ROWS];
    {
        const float w = W1[j], b = B1[j];
#pragma unroll
        for (int r = 0; r < ROWS; ++r) {
            const float t = tanhf(fmaf(w, DT[row0 + r], b));
            y[r]  = t;
            dy[r] = (1.f - t * t) * w;
        }
    }
    const float* Ws[3] = {W2, W3, W4};
    const float* Bs[3] = {B2, B3, B4};
    for (int L = 0; L < 3; ++L) {
#pragma unroll
        for (int r = 0; r < ROWS; ++r) { sa[r][j] = y[r]; sd[r][j] = dy[r]; }
        __syncthreads();
        float z[ROWS], dz[ROWS];
        const float b = Bs[L][j];
#pragma unroll
        for (int r = 0; r < ROWS; ++r) { z[r] = b; dz[r] = 0.f; }
        const float* __restrict__ W = Ws[L];
        for (int i = 0; i < PW; ++i) {
            const float wij = W[i * PW + j];
#pragma unroll
            for (int r = 0; r < ROWS; ++r) {
                z[r]  = fmaf(sa[r][i], wij, z[r]);
                dz[r] = fmaf(sd[r][i], wij, dz[r]);
            }
        }
#pragma unroll
        for (int r = 0; r < ROWS; ++r) {
            const float t = tanhf(z[r]);
            y[r]  = t;
            dy[r] = (1.f - t * t) * dz[r];
        }
        __syncthreads();
    }
    const float wl = wlin[j];
#pragma unroll
    for (int r = 0; r < ROWS; ++r) {
        A  [(size_t)(row0 + r) * PW + j] = y[r]  * wl;
        Amu[(size_t)(row0 + r) * PW + j] = dy[r] * wl;
    }
}

// ---------------------------------------------------------------------------
// Trunk MLP (2 -> 128x4, tanh) forward + JVPs with tangents (1,0), (0,1).
// ---------------------------------------------------------------------------
__global__ __launch_bounds__(PW) void trunk_mlp_kernel(
    const float* __restrict__ coords,
    const float* __restrict__ W1, const float* __restrict__ B1,
    const float* __restrict__ W2, const float* __restrict__ B2,
    const float* __restrict__ W3, const float* __restrict__ B3,
    const float* __restrict__ W4, const float* __restrict__ B4,
    float* __restrict__ Bt, float* __restrict__ Bx, float* __restrict__ By)
{
    __shared__ float sa [ROWS][PW];
    __shared__ float sdx[ROWS][PW];
    __shared__ float sdy[ROWS][PW];
    const int j    = threadIdx.x;
    const int row0 = blockIdx.x * ROWS;

    float y[ROWS], dx[ROWS], dyv[ROWS];
    {
        const float w0 = W1[j], w1 = W1[PW + j], b = B1[j];
#pragma unroll
        for (int r = 0; r < ROWS; ++r) {
            const float cx = coords[(size_t)(row0 + r) * 2 + 0];
            const float cy = coords[(size_t)(row0 + r) * 2 + 1];
            const float t  = tanhf(fmaf(w0, cx, fmaf(w1, cy, b)));
            const float g  = 1.f - t * t;
            y[r] = t; dx[r] = g * w0; dyv[r] = g * w1;
        }
    }
    const float* Ws[3] = {W2, W3, W4};
    const float* Bs[3] = {B2, B3, B4};
    for (int L = 0; L < 3; ++L) {
#pragma unroll
        for (int r = 0; r < ROWS; ++r) {
            sa[r][j] = y[r]; sdx[r][j] = dx[r]; sdy[r][j] = dyv[r];
        }
        __syncthreads();
        float z[ROWS], zx[ROWS], zy[ROWS];
        const float b = Bs[L][j];
#pragma unroll
        for (int r = 0; r < ROWS; ++r) { z[r] = b; zx[r] = 0.f; zy[r] = 0.f; }
        const float* __restrict__ W = Ws[L];
        for (int i = 0; i < PW; ++i) {
            const float wij = W[i * PW + j];
#pragma unroll
            for (int r = 0; r < ROWS; ++r) {
                z[r]  = fmaf(sa [r][i], wij, z[r]);
                zx[r] = fmaf(sdx[r][i], wij, zx[r]);
                zy[r] = fmaf(sdy[r][i], wij, zy[r]);
            }
        }
#pragma unroll
        for (int r = 0; r < ROWS; ++r) {
            const float t = tanhf(z[r]);
            const float g = 1.f - t * t;
            y[r] = t; dx[r] = g * zx[r]; dyv[r] = g * zy[r];
        }
        __syncthreads();
    }
#pragma unroll
    for (int r = 0; r < ROWS; ++r) {
        Bt[(size_t)(row0 + r) * PW + j] = y[r];
        Bx[(size_t)(row0 + r) * PW + j] = dx[r];
        By[(size_t)(row0 + r) * PW + j] = dyv[r];
    }
}

// ---------------------------------------------------------------------------
// Fused rank-128 GEMMs, 32x32 register tile per wave, all 4 outputs.
// Workspace layout (single base): [A | Amu | Bt | Bx | By], 2 MB apart, so
// every fragment load is base + lane_voffset + imm24 (matrix/subtile/k).
// Double-buffered fragments: next chunk's 10 loads issue before the 16 WMMAs
// consuming the current chunk, avoiding per-load s_wait_loadcnt 0.
// Fragment layout (V_WMMA_F32_16X16X4_F32):
//   A 16x4: lanes 0-15 K=k,k+1; lanes 16-31 K=k+2,k+3 (float2 per lane)
//   C/D 16x16 (8 VGPRs): VGPR r -> M=r (lanes 0-15) / M=r+8 (lanes 16-31)
// ---------------------------------------------------------------------------
__global__ __launch_bounds__(256) void fused_gemm_kernel(
    const float* __restrict__ WSBASE,
    float* __restrict__ outU, float* __restrict__ outX,
    float* __restrict__ outY, float* __restrict__ outMu)
{
    const int lane   = threadIdx.x & 31;
    const int wave   = threadIdx.x >> 5;           // 8 waves: 2 (m) x 4 (n)
    const int laneLo = lane & 15;
    const int half   = lane >> 4;                  // upper half-wave: K += 2
    const int mw     = wave >> 2;
    const int nw     = wave & 3;
    const int mBase  = blockIdx.y * 64 + mw * 32;  // 32 rows of b per wave
    const int nBase  = blockIdx.x * 128 + nw * 32; // 32 rows of e per wave

    const char* base = (const char*)WSBASE;
    // per-lane byte offsets into the A-panel and B-panel (k folded in below)
    const int aOff = (mBase + laneLo) * ROWB + half * 8;
    const int bOff = (nBase + laneLo) * ROWB + half * 8;

#define LD(off) (*(const v2f*)(base + (off)))
#define LOAD_FRAGS(dst, kk)                                                   \
    do {                                                                      \
        dst[0] = LD(aOff + (kk));                              /* a  m0 */    \
        dst[1] = LD(aOff + 16 * ROWB + (kk));                  /* a  m1 */    \
        dst[2] = LD(aOff + 1 * MAT_BYTES + (kk));              /* am m0 */    \
        dst[3] = LD(aOff + 1 * MAT_BYTES + 16 * ROWB + (kk));  /* am m1 */    \
        dst[4] = LD(bOff + 2 * MAT_BYTES + (kk));              /* bt n0 */    \
        dst[5] = LD(bOff + 2 * MAT_BYTES + 16 * ROWB + (kk));  /* bt n1 */    \
        dst[6] = LD(bOff + 3 * MAT_BYTES + (kk));              /* bx n0 */    \
        dst[7] = LD(bOff + 3 * MAT_BYTES + 16 * ROWB + (kk));  /* bx n1 */    \
        dst[8] = LD(bOff + 4 * MAT_BYTES + (kk));              /* by n0 */    \
        dst[9] = LD(bOff + 4 * MAT_BYTES + 16 * ROWB + (kk));  /* by n1 */    \
    } while (0)

    v8f accU [2][2] = {};
    v8f accX [2][2] = {};
    v8f accY [2][2] = {};
    v8f accMu[2][2] = {};

    v2f cur[10], nxt[10];
    LOAD_FRAGS(cur, 0);

#pragma unroll 2
    for (int k = 0; k < PW; k += 4) {
        const int knext = ((k + 4) & (PW - 1)) * 4;   // wrap: harmless reload
        LOAD_FRAGS(nxt, knext);
#pragma unroll
        for (int mi = 0; mi < 2; ++mi) {
            const v2f a  = cur[mi];
            const v2f am = cur[2 + mi];
#pragma unroll
            for (int ni = 0; ni < 2; ++ni) {
                const v2f bt = cur[4 + ni];
                const v2f bx = cur[6 + ni];
                const v2f by = cur[8 + ni];
                accU [mi][ni] = __builtin_amdgcn_wmma_f32_16x16x4_f32(
                    false, a,  false, bt, (short)0, accU [mi][ni], false, false);
                accX [mi][ni] = __builtin_amdgcn_wmma_f32_16x16x4_f32(
                    false, a,  false, bx, (short)0, accX [mi][ni], false, false);
                accY [mi][ni] = __builtin_amdgcn_wmma_f32_16x16x4_f32(
                    false, a,  false, by, (short)0, accY [mi][ni], false, false);
                accMu[mi][ni] = __builtin_amdgcn_wmma_f32_16x16x4_f32(
                    false, am, false, bt, (short)0, accMu[mi][ni], false, false);
            }
        }
#pragma unroll
        for (int i = 0; i < 10; ++i) cur[i] = nxt[i];
    }
#undef LOAD_FRAGS
#undef LD

    // store 2x2 subtiles x 4 outputs, non-temporal (write-once 268 MB stream)
#pragma unroll
    for (int mi = 0; mi < 2; ++mi) {
        const int row0 = mBase + mi * 16 + half * 8;
#pragma unroll
        for (int ni = 0; ni < 2; ++ni) {
            const int col = nBase + ni * 16 + laneLo;
#pragma unroll
            for (int r = 0; r < 8; ++r) {
                const size_t idx = (size_t)(row0 + r) * EDIM + col;
                __builtin_nontemporal_store(accU [mi][ni][r], outU  + idx);
                __builtin_nontemporal_store(accX [mi][ni][r], outX  + idx);
                __builtin_nontemporal_store(accY [mi][ni][r], outY  + idx);
                __builtin_nontemporal_store(accMu[mi][ni][r], outMu + idx);
            }
        }
    }
}

extern "C" void kernel_launch(void* const* d_in, const int* in_sizes, int n_in,
                              void* d_out, int out_size, void* d_ws, size_t ws_size,
                              hipStream_t stream) {
    // setup_inputs order: DT, coords, branch_Ws[0..3], branch_bs[0..3],
    //                     trunk_Ws[0..3], trunk_bs[0..3], w_lin
    const float* DT     = (const float*)d_in[0];
    const float* coords = (const float*)d_in[1];
    const float* bW1 = (const float*)d_in[2];
    const float* bW2 = (const float*)d_in[3];
    const float* bW3 = (const float*)d_in[4];
    const float* bW4 = (const float*)d_in[5];
    const float* bB1 = (const float*)d_in[6];
    const float* bB2 = (const float*)d_in[7];
    const float* bB3 = (const float*)d_in[8];
    const float* bB4 = (const float*)d_in[9];
    const float* tW1 = (const float*)d_in[10];
    const float* tW2 = (const float*)d_in[11];
    const float* tW3 = (const float*)d_in[12];
    const float* tW4 = (const float*)d_in[13];
    const float* tB1 = (const float*)d_in[14];
    const float* tB2 = (const float*)d_in[15];
    const float* tB3 = (const float*)d_in[16];
    const float* tB4 = (const float*)d_in[17];
    const float* wlin = (const float*)d_in[18];

    float* ws = (float*)d_ws;
    const size_t S = (size_t)MAT_STRIDE;           // 524288 floats per matrix
    float* A   = ws;
    float* Amu = ws + S;
    float* Bt  = ws + 2 * S;
    float* Bx  = ws + 3 * S;
    float* By  = ws + 4 * S;

    branch_mlp_kernel<<<BDIM / ROWS, PW, 0, stream>>>(
        DT, bW1, bB1, bW2, bB2, bW3, bB3, bW4, bB4, wlin, A, Amu);
    trunk_mlp_kernel<<<EDIM / ROWS, PW, 0, stream>>>(
        coords, tW1, tB1, tW2, tB2, tW3, tB3, tW4, tB4, Bt, Bx, By);

    float* out = (float*)d_out;
    const size_t N = (size_t)BDIM * EDIM;
    fused_gemm_kernel<<<dim3(EDIM / 128, BDIM / 64), 256, 0, stream>>>(
        ws, out, out + N, out + 2 * N, out + 3 * N);
}